// MultiScaleMixtureGLR_37881611550879
// MI455X (gfx1250) — compile-verified
//
#include <hip/hip_runtime.h>

#define HH 384
#define WW 384
#define NN (HH * WW)   // 147456
#define GG 4
#define FF 8
#define SS 9

typedef __attribute__((ext_vector_type(2))) float v2f;
typedef __attribute__((ext_vector_type(8))) float v8f;

// ---------------------------------------------------------------------------
// K1: L2-normalize features along F, project with multiM via WMMA f32 16x16x4.
//     fM[g][n][v] = sum_f (x[g][f][n]/||x||) * M[g][f][v]
// Each wave32 handles a 16-node tile: D[16 nodes x 16 cols] (cols 8..15 are
// zero-padded since M has only 8 columns).
// A layout (32-bit 16x4): lanes 0-15 -> M=0..15, VGPR0=K0, VGPR1=K1;
//                         lanes 16-31 -> VGPR0=K2, VGPR1=K3.
// ---------------------------------------------------------------------------
__global__ void glr_project_wmma(const float* __restrict__ img,
                                 const float* __restrict__ multiM,
                                 float* __restrict__ fM) {
  const int g    = blockIdx.y;
  const int lane = threadIdx.x & 31;
  const int wave = threadIdx.x >> 5;
  const int m    = lane & 15;   // tile-local node row (also B column index)
  const int hi   = lane >> 4;   // K-half selector
  const int n0   = blockIdx.x * 128 + wave * 16;
  const int node = n0 + m;

  const float* xg = img + (size_t)g * FF * NN;   // [f][n]
  const int kA = hi * 2;
  // This lane's 4 features (covers K = {kA,kA+1} for wmma#1, {kA+4,kA+5} for #2).
  float f0 = xg[(size_t)(kA + 0) * NN + node];
  float f1 = xg[(size_t)(kA + 1) * NN + node];
  float f2 = xg[(size_t)(kA + 4) * NN + node];
  float f3 = xg[(size_t)(kA + 5) * NN + node];

  // Full 8-feature sum of squares: lane l and l+16 hold complementary halves.
  float ss = f0 * f0 + f1 * f1 + f2 * f2 + f3 * f3;
  ss += __shfl_xor(ss, 16, 32);
  const float scale = 1.0f / fmaxf(sqrtf(ss), 1e-12f);

  v2f a1 = { f0 * scale, f1 * scale };
  v2f a2 = { f2 * scale, f3 * scale };

  // B (4x16): lanes 0-15 rows {kA,kA+1} col=m; cols >= 8 are zero padding.
  const float* Mg = multiM + g * FF * FF;        // [f][v]
  const float bz = (m < 8) ? 1.0f : 0.0f;
  const int   v  = m & 7;
  v2f b1 = { bz * Mg[(kA + 0) * FF + v], bz * Mg[(kA + 1) * FF + v] };
  v2f b2 = { bz * Mg[(kA + 4) * FF + v], bz * Mg[(kA + 5) * FF + v] };

  v8f c = {};
  c = __builtin_amdgcn_wmma_f32_16x16x4_f32(false, a1, false, b1, (short)0, c, false, false);
  c = __builtin_amdgcn_wmma_f32_16x16x4_f32(false, a2, false, b2, (short)0, c, false, false);

  // D layout: VGPR r, lanes 0-15 -> D[M=r][N=lane]; lanes 16-31 -> D[M=r+8][N=lane-16].
  // Lane holds column v=m for nodes n0 + hi*8 + r. Store fM node-major [g][n][8].
  if (m < 8) {
    float* o = fM + ((size_t)g * NN + n0 + hi * 8) * FF + m;
#pragma unroll
    for (int r = 0; r < 8; ++r) o[(size_t)r * FF] = c[r];
  }
}

__device__ __forceinline__ float dot8(const float4 a0, const float4 a1,
                                      const float4 b0, const float4 b1) {
  return a0.x * b0.x + a0.y * b0.y + a0.z * b0.z + a0.w * b0.w +
         a1.x * b1.x + a1.y * b1.y + a1.z * b1.z + a1.w * b1.w;
}

// ---------------------------------------------------------------------------
// K2: per node, 9-offset stencil gather of fM, edge weight exp(clip(sim)),
//     per-offset weight array (0 for invalid offsets) + node degree.
// ---------------------------------------------------------------------------
__global__ void glr_edge_weights(const float* __restrict__ fM,
                                 float* __restrict__ w,
                                 float* __restrict__ deg) {
  const int n = blockIdx.x * 256 + threadIdx.x;
  const int g = blockIdx.y;
  if (n >= NN) return;
  const int y = n / WW, x = n - y * WW;

  const float4* fmg = (const float4*)(fM + (size_t)g * NN * FF);
  const float4 a0 = fmg[(size_t)n * 2 + 0];
  const float4 a1 = fmg[(size_t)n * 2 + 1];

  float d = 0.0f;
#pragma unroll
  for (int off = 0; off < 9; ++off) {
    const int dy = off / 3 - 1, dx = off % 3 - 1;
    const int jy = y + dy, jx = x + dx;
    const bool ok = (jy >= 0) && (jy < HH) && (jx >= 0) && (jx < WW);
    const int  j  = ok ? (jy * WW + jx) : n;
    const float4 b0 = fmg[(size_t)j * 2 + 0];
    const float4 b1 = fmg[(size_t)j * 2 + 1];
    float sim = dot8(a0, a1, b0, b1);
    sim = fminf(fmaxf(sim, -10.0f), 10.0f);
    const float wv = ok ? expf(sim) : 0.0f;
    d += wv;
    w[((size_t)g * 9 + off) * NN + n] = wv;
  }
  deg[(size_t)g * NN + n] = d;
}

// ---------------------------------------------------------------------------
// K3: symmetric normalization + Laplacian apply:
//     out[s][i] = sig[s][i] - sum_j dinv[i]*w(i,j)*dinv[j]*sig[s][j]
// ---------------------------------------------------------------------------
__global__ void glr_apply(const float* __restrict__ sig,
                          const float* __restrict__ w,
                          const float* __restrict__ deg,
                          float* __restrict__ out) {
  const int n = blockIdx.x * 256 + threadIdx.x;
  const int g = blockIdx.y;
  if (n >= NN) return;
  const int y = n / WW, x = n - y * WW;

  const float* degg   = deg + (size_t)g * NN;
  const float  dinv_i = 1.0f / sqrtf(degg[n]);

  float wn[9];
  int   jj[9];
#pragma unroll
  for (int off = 0; off < 9; ++off) {
    const int dy = off / 3 - 1, dx = off % 3 - 1;
    const int jy = y + dy, jx = x + dx;
    const bool ok = (jy >= 0) && (jy < HH) && (jx >= 0) && (jx < WW);
    const int  j  = ok ? (jy * WW + jx) : n;
    jj[off] = j;
    const float wv = w[((size_t)g * 9 + off) * NN + n];   // 0 when !ok
    wn[off] = dinv_i * wv * (1.0f / sqrtf(degg[j]));
  }

#pragma unroll
  for (int s = 0; s < SS; ++s) {
    const float* sg = sig + ((size_t)g * SS + s) * NN;
    float acc = 0.0f;
#pragma unroll
    for (int off = 0; off < 9; ++off) acc += wn[off] * sg[jj[off]];
    out[((size_t)g * SS + s) * NN + n] = sg[n] - acc;
  }
}

// ---------------------------------------------------------------------------
extern "C" void kernel_launch(void* const* d_in, const int* in_sizes, int n_in,
                              void* d_out, int out_size, void* d_ws, size_t ws_size,
                              hipStream_t stream) {
  const float* img = (const float*)d_in[0];  // [1,G,F,H,W]
  const float* sig = (const float*)d_in[1];  // [1,G,9,N]
  const float* M   = (const float*)d_in[2];  // [G,F,F]
  // d_in[3]/d_in[4] (nodeI/nodeJ) encode the deterministic 3x3 stencil; we
  // exploit the structure directly (gather form, no atomics).

  float* ws  = (float*)d_ws;
  float* fM  = ws;                               // G*N*8  floats (18.9 MB)
  float* w   = fM + (size_t)GG * NN * FF;        // G*9*N  floats (21.2 MB)
  float* deg = w + (size_t)GG * 9 * NN;          // G*N    floats ( 2.4 MB)

  glr_project_wmma<<<dim3(NN / 128, GG), 256, 0, stream>>>(img, M, fM);
  glr_edge_weights<<<dim3(NN / 256, GG), 256, 0, stream>>>(fM, w, deg);
  glr_apply       <<<dim3(NN / 256, GG), 256, 0, stream>>>(sig, w, deg, (float*)d_out);
}